// RGCNEncoder_1262720385450
// MI455X (gfx1250) — compile-verified
//
#include <hip/hip_runtime.h>
#include <hip/hip_bf16.h>

// RGCNConv for MI455X (gfx1250): aggregate-first (scatter-mean per relation),
// then 9 fused [N,64]x[64,64] GEMMs on the f32 WMMA path (V_WMMA_F32_16X16X4_F32).

typedef __attribute__((ext_vector_type(2))) float v2f;  // A/B fragment: 16x4 f32 over 32 lanes
typedef __attribute__((ext_vector_type(8))) float v8f;  // C/D fragment: 16x16 f32 over 32 lanes

#define F 64  // F_IN == F_OUT == 64

// ---------------------------------------------------------------- zero ws
__global__ __launch_bounds__(256) void rgcn_zero(float* __restrict__ p, size_t n) {
    size_t i = ((size_t)blockIdx.x * blockDim.x + threadIdx.x) * 4;
    if (i + 4 <= n) {
        *reinterpret_cast<float4*>(p + i) = make_float4(0.f, 0.f, 0.f, 0.f);
    } else {
        for (; i < n; ++i) p[i] = 0.f;
    }
}

// ------------------------------------------------- per-relation scatter-add
// thread t -> edge e = t/64, feature f = t%64
__global__ __launch_bounds__(256) void rgcn_scatter(
    const float* __restrict__ x, const int* __restrict__ src,
    const int* __restrict__ dst, const int* __restrict__ etype,
    float* __restrict__ agg,   // [R][N][64]
    float* __restrict__ cnt,   // [R][N]
    int n_nodes, int n_edges) {
    long long t = (long long)blockIdx.x * blockDim.x + threadIdx.x;
    long long e = t >> 6;
    int f = (int)(t & 63);
    if (e >= n_edges) return;
    int s = src[e];
    int d = dst[e];
    int r = etype[e];
    float v = x[(size_t)s * F + f];
    atomicAdd(&agg[((size_t)r * n_nodes + d) * F + f], v);
    if (f == 0) atomicAdd(&cnt[(size_t)r * n_nodes + d], 1.0f);
}

// ------------------------------------------------------------ fused GEMMs
// out[i] = x[i]@root + bias + sum_r (agg_r[i]/max(cnt_r[i],1)) @ W[r]
// 8 waves per block; each wave computes one 16-row tile of out (16x64).
__global__ __launch_bounds__(256) void rgcn_gemm(
    const float* __restrict__ x,       // [N][64]
    const float* __restrict__ agg,     // [R][N][64]
    const float* __restrict__ cnt,     // [R][N]
    const float* __restrict__ weight,  // [R][64][64]
    const float* __restrict__ root,    // [64][64]
    const float* __restrict__ bias,    // [64]
    float* __restrict__ out,           // [N][64]
    int n_nodes, int n_rel) {
    __shared__ float sW[F * F];  // 16 KB: current 64x64 weight matrix

    const int tid  = threadIdx.x;
    const int wave = tid >> 5;
    const int lane = tid & 31;
    const int half = lane >> 4;   // 0: lanes 0-15, 1: lanes 16-31
    const int l16  = lane & 15;

    int row_base = ((int)blockIdx.x * 8 + wave) * 16;
    const bool active = (row_base < n_nodes);
    if (!active) row_base = n_nodes - 16;  // clamp: keep loads in-bounds, EXEC all-1s

    // C/D accumulators: acc[nt] covers columns [nt*16, nt*16+16)
    v8f acc[4];
#pragma unroll
    for (int nt = 0; nt < 4; ++nt) {
        float bv = bias[nt * 16 + l16];
#pragma unroll
        for (int i = 0; i < 8; ++i) acc[nt][i] = bv;
    }

    // m == 0: A = x, W = root; m >= 1: A = agg[m-1]/cnt, W = weight[m-1]
    for (int m = 0; m <= n_rel; ++m) {
        const float* gW = (m == 0) ? root : (weight + (size_t)(m - 1) * F * F);
        __syncthreads();
        for (int i = tid; i < F * F; i += 256) sW[i] = gW[i];
        __syncthreads();

        const float* Aptr;
        float scale;
        if (m == 0) {
            Aptr = x;
            scale = 1.0f;
        } else {
            Aptr = agg + (size_t)(m - 1) * n_nodes * F;
            float c = cnt[(size_t)(m - 1) * n_nodes + row_base + l16];
            scale = 1.0f / fmaxf(c, 1.0f);  // per-relation mean (clip(cnt,1))
        }
        const float* arow = Aptr + (size_t)(row_base + l16) * F;

#pragma unroll 4
        for (int kk = 0; kk < 16; ++kk) {
            const int k0 = kk * 4 + half * 2;  // this half-wave's K pair
            v2f a;
            a.x = arow[k0] * scale;
            a.y = arow[k0 + 1] * scale;
#pragma unroll
            for (int nt = 0; nt < 4; ++nt) {
                v2f b;
                b.x = sW[k0 * F + nt * 16 + l16];
                b.y = sW[(k0 + 1) * F + nt * 16 + l16];
                // (neg_a, A, neg_b, B, c_mod, C, reuse_a, reuse_b)
                acc[nt] = __builtin_amdgcn_wmma_f32_16x16x4_f32(
                    false, a, false, b, (short)0, acc[nt], false, false);
            }
        }
    }

    if (active) {
        // D layout: VGPR v -> row M = v + 8*half, col N = nt*16 + l16
#pragma unroll
        for (int nt = 0; nt < 4; ++nt) {
#pragma unroll
            for (int v = 0; v < 8; ++v) {
                out[(size_t)(row_base + v + 8 * half) * F + nt * 16 + l16] = acc[nt][v];
            }
        }
    }
}

extern "C" void kernel_launch(void* const* d_in, const int* in_sizes, int n_in,
                              void* d_out, int out_size, void* d_ws, size_t ws_size,
                              hipStream_t stream) {
    const float* x      = (const float*)d_in[0];
    const int*   eidx   = (const int*)d_in[1];   // [2][E]: row 0 = src, row 1 = dst
    const int*   etype  = (const int*)d_in[2];   // [E]
    const float* weight = (const float*)d_in[3]; // [R][64][64]
    const float* root   = (const float*)d_in[4]; // [64][64]
    const float* bias   = (const float*)d_in[5]; // [64]
    float*       out    = (float*)d_out;

    const int n_edges = in_sizes[2];
    const int n_nodes = in_sizes[0] / F;
    const int n_rel   = in_sizes[3] / (F * F);

    const int* src = eidx;
    const int* dst = eidx + n_edges;

    float* agg = (float*)d_ws;                       // R*N*64 floats
    float* cnt = agg + (size_t)n_rel * n_nodes * F;  // R*N floats

    // 1) zero workspace (atomic accumulators) every call
    const size_t total_f = (size_t)n_rel * n_nodes * F + (size_t)n_rel * n_nodes;
    {
        size_t nthread = (total_f + 3) / 4;
        int blocks = (int)((nthread + 255) / 256);
        rgcn_zero<<<blocks, 256, 0, stream>>>((float*)d_ws, total_f);
    }

    // 2) per-relation scatter-add of source features + in-degree counts
    {
        long long nthread = (long long)n_edges * F;
        int blocks = (int)((nthread + 255) / 256);
        rgcn_scatter<<<blocks, 256, 0, stream>>>(x, src, dst, etype, agg, cnt,
                                                 n_nodes, n_edges);
    }

    // 3) fused mean + 9 GEMMs via f32 WMMA
    {
        int blocks = (n_nodes + 127) / 128;  // 8 waves/block * 16 rows/wave
        rgcn_gemm<<<blocks, 256, 0, stream>>>(x, agg, cnt, weight, root, bias, out,
                                              n_nodes, n_rel);
    }
}